// OriginalGNNStateNet_15616501088755
// MI455X (gfx1250) — compile-verified
//
#include <hip/hip_runtime.h>
#include <math.h>

// ---------------- problem constants (match reference) ----------------
#define NT_C  262144
#define ND_C  262144
#define EDT_C 2097152
#define ETT_C 1048576

typedef float    v2f  __attribute__((ext_vector_type(2)));
typedef float    v8f  __attribute__((ext_vector_type(8)));

__device__ __forceinline__ float lrelu(float x, float s) { return x > 0.f ? x : s * x; }

// float atomic max via monotone int mapping; buffer must be initialized to -inf
__device__ __forceinline__ void atomicMaxF(float* addr, float val) {
  if (val >= 0.f) atomicMax((int*)addr, __float_as_int(val));
  else            atomicMin((unsigned int*)addr, __float_as_uint(val));
}

// ---------------------------------------------------------------------
__global__ void fill4_kernel(float4* __restrict__ p, float v, int n4) {
  int i = blockIdx.x * blockDim.x + threadIdx.x;
  if (i < n4) p[i] = make_float4(v, v, v, v);
}

// out = leaky_relu(LayerNorm(x @ W + b), 0.01), x:[N,F], W:[F,16]
__global__ void stem_kernel(const float* __restrict__ x, const float* __restrict__ W,
                            const float* __restrict__ bias, const float* __restrict__ g,
                            const float* __restrict__ bln, float* __restrict__ out,
                            int N, int F) {
  int n = blockIdx.x * blockDim.x + threadIdx.x;
  if (n >= N) return;
  float acc[16];
  {
    const float4* b4 = (const float4*)bias;
#pragma unroll
    for (int q = 0; q < 4; ++q) {
      float4 t = b4[q];
      acc[4 * q + 0] = t.x; acc[4 * q + 1] = t.y; acc[4 * q + 2] = t.z; acc[4 * q + 3] = t.w;
    }
  }
  const float* xp = x + (size_t)n * F;
  for (int k = 0; k < F; ++k) {
    float xv = xp[k];
    const float4* wk = (const float4*)(W + k * 16);
#pragma unroll
    for (int q = 0; q < 4; ++q) {
      float4 w = wk[q];
      acc[4 * q + 0] += xv * w.x; acc[4 * q + 1] += xv * w.y;
      acc[4 * q + 2] += xv * w.z; acc[4 * q + 3] += xv * w.w;
    }
  }
  float mu = 0.f;
#pragma unroll
  for (int j = 0; j < 16; ++j) mu += acc[j];
  mu *= (1.f / 16.f);
  float var = 0.f;
#pragma unroll
  for (int j = 0; j < 16; ++j) { float d = acc[j] - mu; var += d * d; }
  var *= (1.f / 16.f);
  float rinv = rsqrtf(var + 1e-5f);
  float4* op = (float4*)(out + (size_t)n * 16);
  const float4* g4 = (const float4*)g;
  const float4* bl4 = (const float4*)bln;
#pragma unroll
  for (int q = 0; q < 4; ++q) {
    float4 gg = g4[q], bb = bl4[q], r;
    r.x = lrelu(gg.x * (acc[4 * q + 0] - mu) * rinv + bb.x, 0.01f);
    r.y = lrelu(gg.y * (acc[4 * q + 1] - mu) * rinv + bb.y, 0.01f);
    r.z = lrelu(gg.z * (acc[4 * q + 2] - mu) * rinv + bb.z, 0.01f);
    r.w = lrelu(gg.w * (acc[4 * q + 3] - mu) * rinv + bb.w, 0.01f);
    op[q] = r;
  }
}

// ---------------- WMMA dense transform: Y = X @ W ----------------
// X:[N,16] f32, W:[16, 16*nColTiles] f32, Y:[N, 16*nColTiles] f32.
// One wave per 16-row tile; K=16 done as 4 chained V_WMMA_F32_16X16X4_F32.
__global__ void wmma_xform_kernel(const float* __restrict__ X, const float* __restrict__ W,
                                  float* __restrict__ Y, int nColTiles) {
  const int lane = threadIdx.x & 31;
  const int wave = threadIdx.x >> 5;
  const int tile = blockIdx.x * (blockDim.x >> 5) + wave;
  const int row0 = tile * 16;
  const int r    = lane & 15;   // A: row within tile ; B/D: column within tile
  const int hi   = lane >> 4;   // lane half selects K / M sub-block per ISA layout
  const int Ncols = nColTiles * 16;
  // A fragment: 16x4 f32: lanes 0-15 -> M=lane,K={k0,k0+1}; lanes 16-31 -> K={k0+2,k0+3}
  v2f a[4];
#pragma unroll
  for (int s = 0; s < 4; ++s) {
    float2 t = *(const float2*)(X + (size_t)(row0 + r) * 16 + s * 4 + hi * 2);
    a[s][0] = t.x;
    a[s][1] = t.y;
  }
  for (int ct = 0; ct < nColTiles; ++ct) {
    int col = ct * 16 + r;
    v8f c = {};
#pragma unroll
    for (int s = 0; s < 4; ++s) {
      int k = s * 4 + hi * 2;
      v2f b;
      b[0] = W[(size_t)k * Ncols + col];
      b[1] = W[(size_t)(k + 1) * Ncols + col];
      c = __builtin_amdgcn_wmma_f32_16x16x4_f32(false, a[s], false, b, (short)0, c, false, false);
    }
    // D layout: VGPR i -> M = hi*8 + i, N = lane&15
    float* yp = Y + (size_t)(row0 + hi * 8) * Ncols + ct * 16 + r;
#pragma unroll
    for (int i = 0; i < 8; ++i) yp[(size_t)i * Ncols] = c[i];
  }
}

// ---------------- edge pass 1: score + segment max ----------------
__global__ void edge_score_kernel(const float* __restrict__ xl, const float* __restrict__ xr,
                                  const int* __restrict__ src, const int* __restrict__ dst,
                                  const int* __restrict__ mask, const float* __restrict__ att,
                                  float* __restrict__ sc, float* __restrict__ smax, int E) {
  int e = blockIdx.x * blockDim.x + threadIdx.x;
  if (e >= E) return;
  if (mask && mask[e] == 0) return;   // masked edges contribute nothing
  int s = src[e], d = dst[e];
  const float4* pl = (const float4*)(xl + (size_t)s * 32);   // 8 x B128
  const float4* pr = (const float4*)(xr + (size_t)d * 32);
  const float4* pa = (const float4*)att;
  float acc0 = 0.f, acc1 = 0.f;
#pragma unroll
  for (int q = 0; q < 8; ++q) {       // q<4: head0 ; q>=4: head1
    float4 l = pl[q], rr = pr[q], a4 = pa[q];
    float t = lrelu(l.x + rr.x, 0.2f) * a4.x
            + lrelu(l.y + rr.y, 0.2f) * a4.y
            + lrelu(l.z + rr.z, 0.2f) * a4.z
            + lrelu(l.w + rr.w, 0.2f) * a4.w;
    if (q < 4) acc0 += t; else acc1 += t;
  }
  ((float2*)sc)[e] = make_float2(acc0, acc1);
  atomicMaxF(&smax[2 * (size_t)d],     acc0);
  atomicMaxF(&smax[2 * (size_t)d + 1], acc1);
}

// ---------------- edge pass 2: a = exp(score - smax[dst]); denom += a ----------------
__global__ void edge_exp_kernel(float* __restrict__ sc, const int* __restrict__ dst,
                                const int* __restrict__ mask, const float* __restrict__ smax,
                                float* __restrict__ denom, int E) {
  int e = blockIdx.x * blockDim.x + threadIdx.x;
  if (e >= E) return;
  if (mask && mask[e] == 0) return;
  int d = dst[e];
  float2 s2 = ((const float2*)sc)[e];
  float2 m2 = ((const float2*)smax)[d];
  float a0 = __expf(s2.x - m2.x);
  float a1 = __expf(s2.y - m2.y);
  ((float2*)sc)[e] = make_float2(a0, a1);
  atomicAdd(&denom[2 * (size_t)d],     a0);
  atomicAdd(&denom[2 * (size_t)d + 1], a1);
}

// ---------------- edge pass 3: agg[dst] += mean_h(alpha_h * xl[src,h,:]) ----------------
__global__ void edge_scatter_kernel(const float* __restrict__ a, const int* __restrict__ src,
                                    const int* __restrict__ dst, const int* __restrict__ mask,
                                    const float* __restrict__ denom, const float* __restrict__ xl,
                                    float* __restrict__ agg, int E) {
  int e = blockIdx.x * blockDim.x + threadIdx.x;
  if (e >= E) return;
  if (mask && mask[e] == 0) return;
  int s = src[e], d = dst[e];
  float2 a2 = ((const float2*)a)[e];
  float2 dn = ((const float2*)denom)[d];
  float al0 = 0.5f * a2.x / fmaxf(dn.x, 1e-16f);   // head-mean folded in
  float al1 = 0.5f * a2.y / fmaxf(dn.y, 1e-16f);
  const float4* pl = (const float4*)(xl + (size_t)s * 32);
  float* pd = agg + (size_t)d * 16;
#pragma unroll
  for (int q = 0; q < 4; ++q) {
    float4 l0 = pl[q];       // head0 channels 4q..4q+3
    float4 l1 = pl[4 + q];   // head1
    atomicAdd(&pd[4 * q + 0], al0 * l0.x + al1 * l1.x);
    atomicAdd(&pd[4 * q + 1], al0 * l0.y + al1 * l1.y);
    atomicAdd(&pd[4 * q + 2], al0 * l0.z + al1 * l1.z);
    atomicAdd(&pd[4 * q + 3], al0 * l0.w + al1 * l1.w);
  }
}

// ---------------- finalize: act(LN(agg + x_dst @ resW + bias)) ----------------
__global__ void finalize_kernel(const float* __restrict__ agg, const float* __restrict__ xdst,
                                const float* __restrict__ resW, const float* __restrict__ bias,
                                const float* __restrict__ g, const float* __restrict__ bln,
                                float* __restrict__ out, int N) {
  int n = blockIdx.x * blockDim.x + threadIdx.x;
  if (n >= N) return;
  float o[16], xv[16];
  {
    const float4* xa4 = (const float4*)(xdst + (size_t)n * 16);
    const float4* ag4 = (const float4*)(agg + (size_t)n * 16);
    const float4* b4  = (const float4*)bias;
#pragma unroll
    for (int q = 0; q < 4; ++q) {
      float4 xq = xa4[q], aq = ag4[q], bq = b4[q];
      xv[4 * q + 0] = xq.x; xv[4 * q + 1] = xq.y; xv[4 * q + 2] = xq.z; xv[4 * q + 3] = xq.w;
      o[4 * q + 0] = aq.x + bq.x; o[4 * q + 1] = aq.y + bq.y;
      o[4 * q + 2] = aq.z + bq.z; o[4 * q + 3] = aq.w + bq.w;
    }
  }
#pragma unroll
  for (int k = 0; k < 16; ++k) {
    float xk = xv[k];
    const float4* wr = (const float4*)(resW + k * 16);
#pragma unroll
    for (int q = 0; q < 4; ++q) {
      float4 w = wr[q];
      o[4 * q + 0] += xk * w.x; o[4 * q + 1] += xk * w.y;
      o[4 * q + 2] += xk * w.z; o[4 * q + 3] += xk * w.w;
    }
  }
  float mu = 0.f;
#pragma unroll
  for (int c = 0; c < 16; ++c) mu += o[c];
  mu *= (1.f / 16.f);
  float var = 0.f;
#pragma unroll
  for (int c = 0; c < 16; ++c) { float d = o[c] - mu; var += d * d; }
  var *= (1.f / 16.f);
  float rinv = rsqrtf(var + 1e-5f);
  float4* op = (float4*)(out + (size_t)n * 16);
  const float4* g4 = (const float4*)g;
  const float4* bl4 = (const float4*)bln;
#pragma unroll
  for (int q = 0; q < 4; ++q) {
    float4 gg = g4[q], bb = bl4[q], r;
    r.x = lrelu(gg.x * (o[4 * q + 0] - mu) * rinv + bb.x, 0.01f);
    r.y = lrelu(gg.y * (o[4 * q + 1] - mu) * rinv + bb.y, 0.01f);
    r.z = lrelu(gg.z * (o[4 * q + 2] - mu) * rinv + bb.z, 0.01f);
    r.w = lrelu(gg.w * (o[4 * q + 3] - mu) * rinv + bb.w, 0.01f);
    op[q] = r;
  }
}

// ---------------- pooling: per-block (1024 nodes) LDS reduce, then atomics ----------------
__global__ void pool_stage1(const float* __restrict__ xt, const float* __restrict__ u_tt,
                            const float* __restrict__ u_dt, const int* __restrict__ b_tasks,
                            float* __restrict__ sums, float* __restrict__ cnts) {
  __shared__ float ssum[48];
  if (threadIdx.x < 48) ssum[threadIdx.x] = 0.f;
  __syncthreads();
  size_t base = (size_t)blockIdx.x * 1024;
  float acc[48];
#pragma unroll
  for (int j = 0; j < 48; ++j) acc[j] = 0.f;
  for (int it = 0; it < 4; ++it) {
    size_t n = base + it * 256 + threadIdx.x;
    const float4* p0 = (const float4*)(xt   + n * 16);
    const float4* p1 = (const float4*)(u_tt + n * 16);
    const float4* p2 = (const float4*)(u_dt + n * 16);
#pragma unroll
    for (int q = 0; q < 4; ++q) {
      float4 a = p0[q], b = p1[q], c = p2[q];
      acc[4 * q + 0] += a.x; acc[4 * q + 1] += a.y; acc[4 * q + 2] += a.z; acc[4 * q + 3] += a.w;
      acc[16 + 4 * q + 0] += b.x; acc[16 + 4 * q + 1] += b.y;
      acc[16 + 4 * q + 2] += b.z; acc[16 + 4 * q + 3] += b.w;
      acc[32 + 4 * q + 0] += c.x; acc[32 + 4 * q + 1] += c.y;
      acc[32 + 4 * q + 2] += c.z; acc[32 + 4 * q + 3] += c.w;
    }
  }
#pragma unroll
  for (int j = 0; j < 48; ++j) atomicAdd(&ssum[j], acc[j]);
  __syncthreads();
  int g = b_tasks[base];   // block fully inside one graph (16384 % 1024 == 0)
  if (threadIdx.x < 48) atomicAdd(&sums[g * 48 + threadIdx.x], ssum[threadIdx.x]);
  if (threadIdx.x == 0) atomicAdd(&cnts[g], 1024.f);
}

__global__ void pool_final(const float* __restrict__ xt, const float* __restrict__ u_tt,
                           const float* __restrict__ u_dt, const int* __restrict__ ptr,
                           const float* __restrict__ sums, const float* __restrict__ cnts,
                           float* __restrict__ out) {
  int g = blockIdx.x, j = threadIdx.x;   // grid 16, block 96
  float v;
  if (j < 48) {
    size_t n = (size_t)ptr[g];
    v = (j < 16) ? xt[n * 16 + j]
        : (j < 32) ? u_tt[n * 16 + (j - 16)]
                   : u_dt[n * 16 + (j - 32)];
  } else {
    int k = j - 48;
    v = sums[g * 48 + k] / fmaxf(cnts[g], 1.f);
  }
  out[g * 96 + j] = v;
}

// ---------------------------------------------------------------------
extern "C" void kernel_launch(void* const* d_in, const int* in_sizes, int n_in,
                              void* d_out, int out_size, void* d_ws, size_t ws_size,
                              hipStream_t stream) {
  const float* x_tasks  = (const float*)d_in[0];
  const float* x_data   = (const float*)d_in[1];
  const float* stem_t_W = (const float*)d_in[2];
  const float* stem_t_b = (const float*)d_in[3];
  const float* stem_d_W = (const float*)d_in[4];
  const float* stem_d_b = (const float*)d_in[5];
  const float* ln_t_g   = (const float*)d_in[6];
  const float* ln_t_b   = (const float*)d_in[7];
  const float* ln_d_g   = (const float*)d_in[8];
  const float* ln_d_b   = (const float*)d_in[9];
  const float* dt_Wl    = (const float*)d_in[10];
  const float* dt_Wr    = (const float*)d_in[11];
  const float* dt_att   = (const float*)d_in[12];
  const float* dt_res   = (const float*)d_in[13];
  const float* dt_bias  = (const float*)d_in[14];
  const float* tt_Wl    = (const float*)d_in[15];
  const float* tt_Wr    = (const float*)d_in[16];
  const float* tt_att   = (const float*)d_in[17];
  const float* tt_res   = (const float*)d_in[18];
  const float* tt_bias  = (const float*)d_in[19];
  const float* ln1_g    = (const float*)d_in[20];
  const float* ln1_b    = (const float*)d_in[21];
  const float* ln2_g    = (const float*)d_in[22];
  const float* ln2_b    = (const float*)d_in[23];
  const int*   ei_dt    = (const int*)d_in[24];
  const int*   mask_dt  = (const int*)d_in[25];
  const int*   ei_tt    = (const int*)d_in[26];
  const int*   b_tasks  = (const int*)d_in[27];
  const int*   ptr      = (const int*)d_in[28];
  (void)in_sizes; (void)n_in; (void)out_size;

  // workspace layout (floats); all sub-buffers stay 16B-aligned
  float* ws = (float*)d_ws;
  float* xt     = ws;  ws += (size_t)NT_C * 16;
  float* xd     = ws;  ws += (size_t)ND_C * 16;
  float* xl     = ws;  ws += (size_t)ND_C * 32;
  float* xr     = ws;  ws += (size_t)NT_C * 32;
  float* sc     = ws;  ws += (size_t)EDT_C * 2;
  float* smax   = ws;  ws += (size_t)NT_C * 2;
  float* denom  = ws;  ws += (size_t)NT_C * 2;
  float* agg    = ws;  ws += (size_t)NT_C * 16;
  float* upd_dt = ws;  ws += (size_t)NT_C * 16;   // x_data_upd  (d->t, ln1)
  float* upd_tt = ws;  ws += (size_t)NT_C * 16;   // x_tasks_upd (t->t, ln2)
  float* psum   = ws;  ws += 16 * 48;
  float* pcnt   = ws;  ws += 16;
  size_t needed = (size_t)((char*)ws - (char*)d_ws);
  if (ws_size < needed) return;

  const float NEG_INF = -(float)INFINITY;

  // stems
  stem_kernel<<<NT_C / 256, 256, 0, stream>>>(x_tasks, stem_t_W, stem_t_b, ln_t_g, ln_t_b, xt, NT_C, 12);
  stem_kernel<<<ND_C / 256, 256, 0, stream>>>(x_data,  stem_d_W, stem_d_b, ln_d_g, ln_d_b, xd, ND_C, 5);

  // ======== GATv2: data -> tasks (masked) ========
  wmma_xform_kernel<<<ND_C / 128, 256, 0, stream>>>(xd, dt_Wl, xl, 2);   // [ND,32]
  wmma_xform_kernel<<<NT_C / 128, 256, 0, stream>>>(xt, dt_Wr, xr, 2);   // [NT,32]
  fill4_kernel<<<(NT_C * 2 / 4 + 255) / 256, 256, 0, stream>>>((float4*)smax, NEG_INF, NT_C * 2 / 4);
  fill4_kernel<<<(NT_C * 2 / 4 + 255) / 256, 256, 0, stream>>>((float4*)denom, 0.f, NT_C * 2 / 4);
  fill4_kernel<<<(NT_C * 16 / 4 + 255) / 256, 256, 0, stream>>>((float4*)agg, 0.f, NT_C * 16 / 4);
  edge_score_kernel<<<EDT_C / 256, 256, 0, stream>>>(xl, xr, ei_dt, ei_dt + EDT_C, mask_dt, dt_att, sc, smax, EDT_C);
  edge_exp_kernel<<<EDT_C / 256, 256, 0, stream>>>(sc, ei_dt + EDT_C, mask_dt, smax, denom, EDT_C);
  edge_scatter_kernel<<<EDT_C / 256, 256, 0, stream>>>(sc, ei_dt, ei_dt + EDT_C, mask_dt, denom, xl, agg, EDT_C);
  finalize_kernel<<<NT_C / 256, 256, 0, stream>>>(agg, xt, dt_res, dt_bias, ln1_g, ln1_b, upd_dt, NT_C);

  // ======== GATv2: tasks -> tasks (reuse buffers) ========
  wmma_xform_kernel<<<NT_C / 128, 256, 0, stream>>>(xt, tt_Wl, xl, 2);
  wmma_xform_kernel<<<NT_C / 128, 256, 0, stream>>>(xt, tt_Wr, xr, 2);
  fill4_kernel<<<(NT_C * 2 / 4 + 255) / 256, 256, 0, stream>>>((float4*)smax, NEG_INF, NT_C * 2 / 4);
  fill4_kernel<<<(NT_C * 2 / 4 + 255) / 256, 256, 0, stream>>>((float4*)denom, 0.f, NT_C * 2 / 4);
  fill4_kernel<<<(NT_C * 16 / 4 + 255) / 256, 256, 0, stream>>>((float4*)agg, 0.f, NT_C * 16 / 4);
  edge_score_kernel<<<ETT_C / 256, 256, 0, stream>>>(xl, xr, ei_tt, ei_tt + ETT_C, nullptr, tt_att, sc, smax, ETT_C);
  edge_exp_kernel<<<ETT_C / 256, 256, 0, stream>>>(sc, ei_tt + ETT_C, nullptr, smax, denom, ETT_C);
  edge_scatter_kernel<<<ETT_C / 256, 256, 0, stream>>>(sc, ei_tt, ei_tt + ETT_C, nullptr, denom, xl, agg, ETT_C);
  finalize_kernel<<<NT_C / 256, 256, 0, stream>>>(agg, xt, tt_res, tt_bias, ln2_g, ln2_b, upd_tt, NT_C);

  // ======== pooling + head ========
  fill4_kernel<<<((16 * 48 + 16) / 4 + 255) / 256, 256, 0, stream>>>((float4*)psum, 0.f, (16 * 48 + 16) / 4);
  pool_stage1<<<NT_C / 1024, 256, 0, stream>>>(xt, upd_tt, upd_dt, b_tasks, psum, pcnt);
  pool_final<<<16, 96, 0, stream>>>(xt, upd_tt, upd_dt, ptr, psum, pcnt, (float*)d_out);
}